// CollisionLoss_31250182045715
// MI455X (gfx1250) — compile-verified
//
#include <hip/hip_runtime.h>
#include <hip/hip_bf16.h>
#include <math.h>

#define BLOCK 256
#define MAXT  16
#define W_EGO 2.35f   /* 1.85 + 0.5 */
#define L_EGO 4.584f  /* 4.084 + 0.5 */

// ---- CDNA5 async global->LDS path (gfx1250). Guarded so compile never fails.
#if defined(__has_builtin)
#  if __has_builtin(__builtin_amdgcn_global_load_async_to_lds_b128)
#    define HAVE_ASYNC_LDS 1
#  endif
#endif
#ifndef HAVE_ASYNC_LDS
#  define HAVE_ASYNC_LDS 0
#endif

typedef int v4i __attribute__((ext_vector_type(4)));
typedef __attribute__((address_space(1))) v4i gv4i;   // global int4
typedef __attribute__((address_space(3))) v4i lv4i;   // LDS int4

__device__ __forceinline__ void wait_async_zero() {
#if HAVE_ASYNC_LDS
#  if defined(__has_builtin) && __has_builtin(__builtin_amdgcn_s_wait_asynccnt)
  __builtin_amdgcn_s_wait_asynccnt(0);
#  else
  asm volatile("s_wait_asynccnt 0x0" ::: "memory");
#  endif
#endif
}

// Faithful _circle_feats, including the buggy edge "distance" = |dx + dy|.
__device__ __forceinline__ void circle_feats(const float cx[4], const float cy[4],
                                             float gx[5], float gy[5], float& width) {
  // width = min_k | (c_k - c_{k+1}).x + (c_k - c_{k+1}).y |
  float wmin = 3.4e38f;
#pragma unroll
  for (int k = 0; k < 4; ++k) {
    const int kn = (k + 1) & 3;
    const float s = (cx[k] - cx[kn]) + (cy[k] - cy[kn]);
    wmin = fminf(wmin, fabsf(s));
  }
  // e_k = c_k - c_{k-1}; length = max |e_k|; idx = FIRST argmax (jnp.argmax)
  float ex[4], ey[4], el[4];
#pragma unroll
  for (int k = 0; k < 4; ++k) {
    const int kp = (k + 3) & 3;
    ex[k] = cx[k] - cx[kp];
    ey[k] = cy[k] - cy[kp];
    el[k] = sqrtf(ex[k] * ex[k] + ey[k] * ey[k]);
  }
  int idx = 0; float best = el[0];
#pragma unroll
  for (int k = 1; k < 4; ++k) { if (el[k] > best) { best = el[k]; idx = k; } }
  const float slope = atanf(ey[idx] / ex[idx]);   // reference uses arctan(y/x), not atan2
  const float mx = (cx[0] + cx[1] + cx[2] + cx[3]) * 0.25f;
  const float my = (cy[0] + cy[1] + cy[2] + cy[3]) * 0.25f;
  const float half = 0.5f * best - 0.5f * wmin;
  const float dx = cosf(slope), dy = sinf(slope);
  const float offs[5] = {0.f, half, -half, 0.5f * half, -0.5f * half};
#pragma unroll
  for (int j = 0; j < 5; ++j) { gx[j] = mx + offs[j] * dx; gy[j] = my + offs[j] * dy; }
  width = wmin;
}

__global__ __launch_bounds__(BLOCK) void collision_main(
    const float* __restrict__ traj,        // [T,2]  (sdc_traj_all[0])
    const float* __restrict__ plan,        // [T,3]  (sdc_planning_gt[0])
    const float* __restrict__ gtc,         // [T*N, 4, 2]
    const unsigned char* __restrict__ gmask,// [T*N] bool
    float* __restrict__ bsum,              // [gridDim.x] partial sums
    int T, int N, int total) {
  __shared__ float s_ecx[MAXT][5];
  __shared__ float s_ecy[MAXT][5];
  __shared__ float s_ew[MAXT];
  __shared__ float s_red[BLOCK];
#if HAVE_ASYNC_LDS
  __shared__ float4 s_buf[BLOCK * 2];      // 8 KB staging tile
#endif

  const int tid = threadIdx.x;
  const int g   = blockIdx.x * BLOCK + tid;
  const bool valid = (g < total);

#if HAVE_ASYNC_LDS
  // Kick off the async DMA of this block's 8KB corner tile into LDS.
  if (valid) {
    const float4* src = ((const float4*)gtc) + (size_t)g * 2;
    gv4i* gp = (gv4i*)src;
    lv4i* lp = (lv4i*)(&s_buf[tid * 2]);
    __builtin_amdgcn_global_load_async_to_lds_b128(gp, lp, 0, 0);
    __builtin_amdgcn_global_load_async_to_lds_b128(gp, lp, 16, 0);
  }
#endif

  // Overlapped with the DMA: first T threads compute ego circle features.
  if (tid < T && tid < MAXT) {
    const float x  = traj[2 * tid + 0];
    const float y  = traj[2 * tid + 1];
    const float th = plan[3 * tid + 2];
    const float c = cosf(th), s = sinf(th);
    const float hw = 0.5f * W_EGO, hl = 0.5f * L_EGO;
    const float lx[4] = {hw, hw, -hw, -hw};
    const float ly[4] = {-hl, hl, hl, -hl};
    float cx[4], cy[4];
#pragma unroll
    for (int k = 0; k < 4; ++k) {
      cx[k] =  c * lx[k] + s * ly[k] + x;
      cy[k] = -s * lx[k] + c * ly[k] + y;
    }
    float exd[5], eyd[5], w;
    circle_feats(cx, cy, exd, eyd, w);
#pragma unroll
    for (int j = 0; j < 5; ++j) { s_ecx[tid][j] = exd[j]; s_ecy[tid][j] = eyd[j]; }
    s_ew[tid] = w;
  }

  wait_async_zero();
  __syncthreads();

  float pen = 0.f;
  if (valid) {
    float4 A, B;
#if HAVE_ASYNC_LDS
    A = s_buf[tid * 2 + 0];
    B = s_buf[tid * 2 + 1];
#else
    const float4* src = ((const float4*)gtc) + (size_t)g * 2;
    A = src[0];
    B = src[1];
#endif
    const float cx[4] = {A.x, A.z, B.x, B.z};
    const float cy[4] = {A.y, A.w, B.y, B.w};
    float gx[5], gy[5], gw;
    circle_feats(cx, cy, gx, gy, gw);

    const int t = g / N;
    // min over 25 sqrt distances == sqrt of min squared distance (sqrt monotone, IEEE).
    float md2 = 3.4e38f;
#pragma unroll
    for (int i = 0; i < 5; ++i) {
      const float sx = s_ecx[t][i], sy = s_ecy[t][i];
#pragma unroll
      for (int j = 0; j < 5; ++j) {
        const float dx = sx - gx[j];
        const float dy = sy - gy[j];
        md2 = fminf(md2, dx * dx + dy * dy);
      }
    }
    const float md = sqrtf(md2);
    pen = fmaxf(0.5f * (s_ew[t] + gw) - md, 0.f);
    pen *= (gmask[g] != 0) ? 1.f : 0.f;
  }

  // Deterministic block tree reduction (replay-stable; no atomics).
  s_red[tid] = pen;
  __syncthreads();
#pragma unroll
  for (int s = BLOCK / 2; s > 0; s >>= 1) {
    if (tid < s) s_red[tid] += s_red[tid + s];
    __syncthreads();
  }
  if (tid == 0) bsum[blockIdx.x] = s_red[0];
}

__global__ __launch_bounds__(BLOCK) void collision_reduce(
    const float* __restrict__ bsum, int nblk, float* __restrict__ out) {
  __shared__ float s_red[BLOCK];
  float acc = 0.f;
  for (int i = threadIdx.x; i < nblk; i += BLOCK) acc += bsum[i];
  s_red[threadIdx.x] = acc;
  __syncthreads();
#pragma unroll
  for (int s = BLOCK / 2; s > 0; s >>= 1) {
    if (threadIdx.x < s) s_red[threadIdx.x] += s_red[threadIdx.x + s];
    __syncthreads();
  }
  if (threadIdx.x == 0) out[0] = s_red[0] * 1.0f;  // WEIGHT = 1.0
}

extern "C" void kernel_launch(void* const* d_in, const int* in_sizes, int n_in,
                              void* d_out, int out_size, void* d_ws, size_t ws_size,
                              hipStream_t stream) {
  const float* traj          = (const float*)d_in[0];          // (1,T,2) f32
  const float* plan          = (const float*)d_in[1];          // (1,T,3) f32
  /* d_in[2] = sdc_planning_gt_mask: unused by the reference loss */
  const float* gt_corners    = (const float*)d_in[3];          // (T,N,4,2) f32
  const unsigned char* gmask = (const unsigned char*)d_in[4];  // (T,N) bool

  const int T = in_sizes[0] / 2;
  const int N = (T > 0) ? (in_sizes[4] / T) : 0;
  const int total = T * N;
  const int nblk = (total + BLOCK - 1) / BLOCK;

  float* bsum = (float*)d_ws;  // nblk floats of scratch, fully rewritten each call

  collision_main<<<nblk, BLOCK, 0, stream>>>(traj, plan, gt_corners, gmask, bsum, T, N, total);
  collision_reduce<<<1, BLOCK, 0, stream>>>(bsum, nblk, (float*)d_out);
}